// GCN_65807488909364
// MI455X (gfx1250) — compile-verified
//
#include <hip/hip_runtime.h>
#include <math.h>

#define HDIM 16

typedef __attribute__((ext_vector_type(2))) float v2f;
typedef __attribute__((ext_vector_type(8))) float v8f;

// ---------------- degree / norm prep ----------------
__global__ void __launch_bounds__(256)
init_kernel(float* deg, float* red, int n) {
    int i = blockIdx.x * blockDim.x + threadIdx.x;
    if (i < n)  deg[i] = 1.0f;   // self-loop weight 1 folded into degree
    if (i < 64) red[i] = 0.0f;   // reduction scratch (mean-pool accumulators etc.)
}

__global__ void __launch_bounds__(256)
deg_kernel(const int* __restrict__ dst, const float* __restrict__ w,
           float* deg, int e) {
    int i = blockIdx.x * blockDim.x + threadIdx.x;
    if (i < e) atomicAdd(&deg[dst[i]], w[i]);
}

__global__ void __launch_bounds__(256)
dinv_kernel(float* deg, int n) {
    int i = blockIdx.x * blockDim.x + threadIdx.x;
    if (i < n) {
        float d = deg[i];
        deg[i] = (d > 0.0f) ? rsqrtf(d) : 0.0f;   // in-place deg -> d^-1/2
    }
}

// ---------------- WMMA fp32 GEMM: out[n x 16] = A[n x K] @ B[K x 16] ----------------
// One wave32 per 16-row tile. n must be a multiple of 16 (100000 = 6250 * 16).
// A-fragment layout (32-bit A 16x4): lanes 0-15 hold K = k,k+1 ; lanes 16-31 hold K = k+2,k+3.
// B-fragment mirrored: lane&15 = output column. C/D: VGPR r = row r (lanes 0-15) / row 8+r (lanes 16-31).
__global__ void __launch_bounds__(256)
gemm_wmma_kernel(const float* __restrict__ A, const float* __restrict__ B,
                 const float* __restrict__ bias, float* __restrict__ out,
                 int n, int K, int relu) {
    int wave   = blockIdx.x * (blockDim.x >> 5) + (threadIdx.x >> 5);
    int ntiles = n >> 4;
    if (wave >= ntiles) return;           // uniform per wave: EXEC stays all-ones for WMMA
    int  lane = threadIdx.x & 31;
    int  m    = lane & 15;                // A row within tile / B,C column
    int  kh   = (lane >> 4) << 1;         // 0 (lanes 0-15) or 2 (lanes 16-31)
    long row0 = (long)wave << 4;
    const float* arow = A + (row0 + m) * K;

    v8f acc = {};
    for (int k = 0; k < K; k += 4) {
        v2f a, b;
        a.x = arow[k + kh];
        a.y = arow[k + kh + 1];
        b.x = B[(k + kh) * HDIM + m];
        b.y = B[(k + kh + 1) * HDIM + m];
        acc = __builtin_amdgcn_wmma_f32_16x16x4_f32(
            /*neg_a=*/false, a, /*neg_b=*/false, b,
            /*c_mod=*/(short)0, acc, /*reuse_a=*/false, /*reuse_b=*/false);
    }

    int   radd = (lane >> 4) << 3;        // lanes 16-31 hold rows 8..15
    float bv   = bias ? bias[m] : 0.0f;
#pragma unroll
    for (int r = 0; r < 8; ++r) {
        float v = acc[r] + bv;
        if (relu) v = fmaxf(v, 0.0f);
        out[(row0 + r + radd) * HDIM + m] = v;
    }
}

// ---------------- aggregation (self-loop init + edge scatter + bias/relu) ----------------
__global__ void __launch_bounds__(256)
selfloop_kernel(const float* __restrict__ hw, const float* __restrict__ dinv,
                float* __restrict__ agg, int n) {
    int idx = blockIdx.x * blockDim.x + threadIdx.x;
    if (idx < n * HDIM) {
        float di = dinv[idx >> 4];
        agg[idx] = hw[idx] * di * di;     // norm_selfloop = dinv[i]*1*dinv[i]
    }
}

__global__ void __launch_bounds__(256)
scatter_kernel(const int* __restrict__ src, const int* __restrict__ dst,
               const float* __restrict__ w, const float* __restrict__ dinv,
               const float* __restrict__ hw, float* __restrict__ agg, int e) {
    long gid = (long)blockIdx.x * blockDim.x + threadIdx.x;
    if (gid >= (long)e * HDIM) return;
    int eidx = (int)(gid >> 4);
    int f    = (int)(gid & 15);
    int s = src[eidx];
    int d = dst[eidx];
    float coef = dinv[s] * w[eidx] * dinv[d];
    atomicAdd(&agg[(long)d * HDIM + f], hw[(long)s * HDIM + f] * coef);
}

__global__ void __launch_bounds__(256)
biasrelu_kernel(const float* __restrict__ agg, const float* __restrict__ b,
                float* __restrict__ h, int n) {
    int idx = blockIdx.x * blockDim.x + threadIdx.x;
    if (idx < n * HDIM)
        h[idx] = fmaxf(agg[idx] + b[idx & 15], 0.0f);
}

// ---------------- head: score, softmax over nodes, mean-pool value ----------------
__global__ void __launch_bounds__(256)
score_kernel(const float* __restrict__ h4, const float* __restrict__ Wl2,
             const float* __restrict__ bl2, float* __restrict__ c, int n) {
    int i = blockIdx.x * blockDim.x + threadIdx.x;
    if (i < n) {
        float v = bl2[0];
#pragma unroll
        for (int f = 0; f < HDIM; ++f) v += h4[(long)i * HDIM + f] * Wl2[f];
        c[i] = v;
    }
}

__global__ void __launch_bounds__(1024)
reduce_kernel(const float* __restrict__ c, float* red, int n) {
    __shared__ float sdata[1024];
    int tid = threadIdx.x;
    float m = -INFINITY;
    for (int i = tid; i < n; i += blockDim.x) m = fmaxf(m, c[i]);
    sdata[tid] = m; __syncthreads();
    for (int s = blockDim.x >> 1; s > 0; s >>= 1) {
        if (tid < s) sdata[tid] = fmaxf(sdata[tid], sdata[tid + s]);
        __syncthreads();
    }
    float gmax = sdata[0];
    __syncthreads();
    float sum = 0.0f;
    for (int i = tid; i < n; i += blockDim.x) sum += expf(c[i] - gmax);
    sdata[tid] = sum; __syncthreads();
    for (int s = blockDim.x >> 1; s > 0; s >>= 1) {
        if (tid < s) sdata[tid] += sdata[tid + s];
        __syncthreads();
    }
    if (tid == 0) { red[0] = gmax; red[1] = sdata[0]; }
}

__global__ void __launch_bounds__(256)
softmax_kernel(const float* __restrict__ c, const float* __restrict__ red,
               float* __restrict__ out, int n) {
    int i = blockIdx.x * blockDim.x + threadIdx.x;
    if (i < n) out[i] = expf(c[i] - red[0]) / red[1];
}

__global__ void __launch_bounds__(256)
meanpool_kernel(const float* __restrict__ h, float* red, int n) {
    __shared__ float fs[HDIM];
    if (threadIdx.x < HDIM) fs[threadIdx.x] = 0.0f;
    __syncthreads();
    int total  = n * HDIM;
    int stride = gridDim.x * blockDim.x;      // multiple of 16 -> f fixed per thread
    int f      = threadIdx.x & 15;
    float acc  = 0.0f;
    for (int idx = blockIdx.x * blockDim.x + threadIdx.x; idx < total; idx += stride)
        acc += h[idx];
    atomicAdd(&fs[f], acc);
    __syncthreads();
    if (threadIdx.x < HDIM) atomicAdd(&red[2 + threadIdx.x], fs[threadIdx.x]);
}

__global__ void
value_kernel(const float* __restrict__ red, const float* __restrict__ Wl2,
             const float* __restrict__ bl2, float* __restrict__ out, int n) {
    if (blockIdx.x == 0 && threadIdx.x == 0) {
        float v = bl2[0];
        float inv_n = 1.0f / (float)n;
#pragma unroll
        for (int f = 0; f < HDIM; ++f) v += (red[2 + f] * inv_n) * Wl2[f];
        out[n] = v;
    }
}

// ---------------- driver ----------------
extern "C" void kernel_launch(void* const* d_in, const int* in_sizes, int n_in,
                              void* d_out, int out_size, void* d_ws, size_t ws_size,
                              hipStream_t stream) {
    const float* x    = (const float*)d_in[0];
    const int*   eidx = (const int*)  d_in[1];   // [2, E] row-major
    const float* w    = (const float*)d_in[2];
    const float* W1   = (const float*)d_in[3];
    const float* b1   = (const float*)d_in[4];
    const float* W2   = (const float*)d_in[5];
    const float* b2   = (const float*)d_in[6];
    const float* W3   = (const float*)d_in[7];
    const float* b3   = (const float*)d_in[8];
    const float* Wl1  = (const float*)d_in[9];
    const float* bl1  = (const float*)d_in[10];
    const float* Wl2  = (const float*)d_in[11];
    const float* bl2  = (const float*)d_in[12];

    const int F = 128;
    int n = in_sizes[0] / F;
    int e = in_sizes[2];
    const int* src = eidx;
    const int* dst = eidx + e;

    float* ws   = (float*)d_ws;
    float* dinv = ws;                          // [n]  (deg, then d^-1/2 in place)
    float* cbuf = ws + n;                      // [n]  per-node score
    float* red  = ws + 2 * (size_t)n;          // [64] reduction scratch
    float* bufA = red + 64;                    // [n*16]
    float* bufB = bufA + (size_t)n * HDIM;     // [n*16]
    float* bufC = bufB + (size_t)n * HDIM;     // [n*16]
    float* out  = (float*)d_out;

    dim3 blk(256);
    int gN  = (n + 255) / 256;
    int gNH = (n * HDIM + 255) / 256;
    int gE  = (e + 255) / 256;
    int gS  = (int)(((long)e * HDIM + 255) / 256);
    int gT  = (n / 16 + 7) / 8;                // 8 wave32 tiles per 256-thread block

    // degree + symmetric normalization
    init_kernel<<<gN, blk, 0, stream>>>(dinv, red, n);
    deg_kernel <<<gE, blk, 0, stream>>>(dst, w, dinv, e);
    dinv_kernel<<<gN, blk, 0, stream>>>(dinv, n);

    // GCN layer 1 (K = 128)
    gemm_wmma_kernel<<<gT, blk, 0, stream>>>(x, W1, nullptr, bufB, n, F, 0);
    selfloop_kernel <<<gNH, blk, 0, stream>>>(bufB, dinv, bufC, n);
    scatter_kernel  <<<gS, blk, 0, stream>>>(src, dst, w, dinv, bufB, bufC, e);
    biasrelu_kernel <<<gNH, blk, 0, stream>>>(bufC, b1, bufA, n);

    // GCN layer 2 (K = 16)
    gemm_wmma_kernel<<<gT, blk, 0, stream>>>(bufA, W2, nullptr, bufB, n, HDIM, 0);
    selfloop_kernel <<<gNH, blk, 0, stream>>>(bufB, dinv, bufC, n);
    scatter_kernel  <<<gS, blk, 0, stream>>>(src, dst, w, dinv, bufB, bufC, e);
    biasrelu_kernel <<<gNH, blk, 0, stream>>>(bufC, b2, bufA, n);

    // GCN layer 3 (K = 16)
    gemm_wmma_kernel<<<gT, blk, 0, stream>>>(bufA, W3, nullptr, bufB, n, HDIM, 0);
    selfloop_kernel <<<gNH, blk, 0, stream>>>(bufB, dinv, bufC, n);
    scatter_kernel  <<<gS, blk, 0, stream>>>(src, dst, w, dinv, bufB, bufC, e);
    biasrelu_kernel <<<gNH, blk, 0, stream>>>(bufC, b3, bufA, n);

    // dense head: h4 = relu(h3 @ Wl1 + bl1)  (bias+relu fused into WMMA store)
    gemm_wmma_kernel<<<gT, blk, 0, stream>>>(bufA, Wl1, bl1, bufB, n, HDIM, 1);

    // choice = softmax over nodes of (h4 @ Wl2 + bl2)
    score_kernel  <<<gN, blk, 0, stream>>>(bufB, Wl2, bl2, cbuf, n);
    reduce_kernel <<<1, 1024, 0, stream>>>(cbuf, red, n);
    softmax_kernel<<<gN, blk, 0, stream>>>(cbuf, red, out, n);

    // value = mean(h4, axis=0) @ Wl2 + bl2
    meanpool_kernel<<<512, blk, 0, stream>>>(bufB, red, n);
    value_kernel   <<<1, 1, 0, stream>>>(red, Wl2, bl2, out, n);
}